// NegativeBinomialRegressionModel_79620103733400
// MI455X (gfx1250) — compile-verified
//
#include <hip/hip_runtime.h>
#include <hip/hip_bf16.h>
#include <math.h>

// Problem shape (from reference)
#define SAMPLES 512
#define GENES   20000
#define COVS    4
#define STILE   16
#define GTILE   16
#define NSTILES (SAMPLES / STILE)   // 32
#define NGTILES (GENES / GTILE)     // 1250
#define GRIDY   16                  // gene-direction block count

typedef __attribute__((ext_vector_type(2))) float v2f;
typedef __attribute__((ext_vector_type(8))) float v8f;

__device__ __forceinline__ float softplusf(float x) {
  return (x > 20.0f) ? x : log1pf(expf(x));
}

// ---------------------------------------------------------------------------
// Pass A: per-sample softmax stats (row max m, sum-exp S) + library size.
// One block (256 threads) per sample; online softmax per thread, LDS tree
// combine. Writes ws[0..511]=m, ws[512..1023]=S, ws[1024..1535]=lib.
// ---------------------------------------------------------------------------
__global__ __launch_bounds__(256)
void nb_stats_kernel(const float* __restrict__ X,
                     const int* __restrict__ Y,
                     const float* __restrict__ beta,
                     float* __restrict__ ws) {
  __shared__ float sm[256], ss[256], sl[256];
  const int n = blockIdx.x;
  const int tid = threadIdx.x;

  const float x0 = X[n * COVS + 0];
  const float x1 = X[n * COVS + 1];
  const float x2 = X[n * COVS + 2];
  const float x3 = X[n * COVS + 3];

  float m_t = -INFINITY, s_t = 0.0f, l_t = 0.0f;
  for (int j = tid; j < GENES; j += 256) {
    float z = x0 * beta[0 * GENES + j] + x1 * beta[1 * GENES + j] +
              x2 * beta[2 * GENES + j] + x3 * beta[3 * GENES + j];
    l_t += (float)Y[(long)n * GENES + j];
    if (z > m_t) {
      s_t = s_t * expf(m_t - z) + 1.0f;  // expf(-INF)==0 on first hit
      m_t = z;
    } else {
      s_t += expf(z - m_t);
    }
  }
  sm[tid] = m_t; ss[tid] = s_t; sl[tid] = l_t;
  __syncthreads();
  for (int str = 128; str > 0; str >>= 1) {
    if (tid < str) {
      float m1 = sm[tid], m2 = sm[tid + str];
      float mm = fmaxf(m1, m2);
      ss[tid] = ss[tid] * expf(m1 - mm) + ss[tid + str] * expf(m2 - mm);
      sm[tid] = mm;
      sl[tid] += sl[tid + str];
    }
    __syncthreads();
  }
  if (tid == 0) {
    ws[n]               = sm[0];  // row max
    ws[SAMPLES + n]     = ss[0];  // sum exp
    ws[2 * SAMPLES + n] = sl[0];  // library size
  }
}

// ---------------------------------------------------------------------------
// Pass B: NB log-likelihood. Each wave computes z for a 16x16 (samples x
// genes) tile with V_WMMA_F32_16X16X4_F32, then evaluates the per-element
// NB terms straight out of the accumulator layout. Per-block partial sums
// go to ws (deterministic; no float atomics).
//
// WMMA layouts (ISA 7.12.2, wave32):
//   A 16x4 f32 : lane l -> M = l&15 ; VGPR{0,1} = K = {2*(l>>4), 2*(l>>4)+1}
//   B 4x16 f32 : lane l -> N = l&15 ; VGPR{0,1} = K = {2*(l>>4), 2*(l>>4)+1}
//   C/D 16x16  : lane l -> N = l&15 ; VGPR g -> M = g + 8*(l>>4)
// ---------------------------------------------------------------------------
__global__ __launch_bounds__(256)
void nb_loglik_kernel(const float* __restrict__ X,
                      const int* __restrict__ Y,
                      const float* __restrict__ beta,
                      const float* __restrict__ phi,
                      const float* __restrict__ ws,
                      float* __restrict__ partials) {
  __shared__ float sx[STILE][COVS];  // X tile (16 samples x 4 covariates)
  __shared__ float srm[STILE];       // row max per sample in tile
  __shared__ float sA[STILE];        // lib[n] / S[n] per sample in tile
  __shared__ float red[256];

  const int st  = blockIdx.x;        // sample tile
  const int tid = threadIdx.x;

  if (tid < STILE * COVS)
    sx[tid / COVS][tid % COVS] = X[(st * STILE + tid / COVS) * COVS + tid % COVS];
  if (tid < STILE) {
    int n = st * STILE + tid;
    srm[tid] = ws[n];
    sA[tid]  = ws[2 * SAMPLES + n] / ws[SAMPLES + n];
  }
  __syncthreads();

  const int lane = tid & 31;
  const int wave = tid >> 5;
  const int hi   = lane >> 4;   // half-wave selects K pair {0,1} vs {2,3}
  const int col  = lane & 15;   // M for A-frag, N (gene col) for B/C frags
  const int kb   = 2 * hi;

#if defined(__gfx1250__) && __has_builtin(__builtin_amdgcn_wmma_f32_16x16x4_f32)
  v2f afrag;
  afrag.x = sx[col][kb + 0];
  afrag.y = sx[col][kb + 1];
#endif

  float acc = 0.0f;
  const int wgid   = blockIdx.y * 8 + wave;  // global wave id in gene dim
  const int wcount = GRIDY * 8;              // 128 waves striding gene tiles

  for (int jt = wgid; jt < NGTILES; jt += wcount) {   // uniform per wave
    const int j = jt * GTILE + col;                   // this lane's gene

    // z tile: D = A(16x4) x B(4x16)
#if defined(__gfx1250__) && __has_builtin(__builtin_amdgcn_wmma_f32_16x16x4_f32)
    v2f bfrag;
    bfrag.x = beta[(long)(kb + 0) * GENES + j];
    bfrag.y = beta[(long)(kb + 1) * GENES + j];
    v8f c = {};
    c = __builtin_amdgcn_wmma_f32_16x16x4_f32(
        /*neg_a=*/false, afrag, /*neg_b=*/false, bfrag,
        /*c_mod=*/(short)0, c, /*reuse_a=*/false, /*reuse_b=*/false);
#else
    float c[8];
    #pragma unroll
    for (int g = 0; g < 8; ++g) {
      int row = g + 8 * hi;
      c[g] = sx[row][0] * beta[(long)0 * GENES + j] +
             sx[row][1] * beta[(long)1 * GENES + j] +
             sx[row][2] * beta[(long)2 * GENES + j] +
             sx[row][3] * beta[(long)3 * GENES + j];
    }
#endif

    // per-gene invariants (hoisted): dispersion d, r = 1/d, lgamma(r)
    const float d   = softplusf(phi[j]);
    const float r   = 1.0f / d;
    const float lgr = lgammaf(r);

    #pragma unroll
    for (int g = 0; g < 8; ++g) {
      const int row = g + 8 * hi;
      const int n   = st * STILE + row;
      const float z  = c[g];
      const float mu = sA[row] * expf(z - srm[row]);   // s * softmax
      const float t  = d * mu;                          // phi*mu
      const float y  = (float)Y[(long)n * GENES + (jt * GTILE + col)];
      const float l1pt = log1pf(t);
      // gammaln(y+r) - gammaln(r) - gammaln(y+1) + r*log(p) + y*log1p(-p)
      //   log(p) = -log1p(t) ; log1p(-p) = log(t) - log1p(t)
      acc += lgammaf(y + r) - lgr - lgammaf(y + 1.0f)
             - r * l1pt + y * (logf(t) - l1pt);
    }
  }

  red[tid] = acc;
  __syncthreads();
  for (int str = 128; str > 0; str >>= 1) {
    if (tid < str) red[tid] += red[tid + str];
    __syncthreads();
  }
  if (tid == 0) partials[st * GRIDY + blockIdx.y] = red[0];
}

// ---------------------------------------------------------------------------
// Pass C: deterministic final reduction of the 512 block partials.
// ---------------------------------------------------------------------------
__global__ __launch_bounds__(256)
void nb_final_kernel(const float* __restrict__ partials, int np,
                     float* __restrict__ out) {
  __shared__ float red[256];
  float a = 0.0f;
  for (int i = threadIdx.x; i < np; i += 256) a += partials[i];
  red[threadIdx.x] = a;
  __syncthreads();
  for (int s = 128; s > 0; s >>= 1) {
    if (threadIdx.x < s) red[threadIdx.x] += red[threadIdx.x + s];
    __syncthreads();
  }
  if (threadIdx.x == 0) out[0] = red[0];
}

extern "C" void kernel_launch(void* const* d_in, const int* in_sizes, int n_in,
                              void* d_out, int out_size, void* d_ws, size_t ws_size,
                              hipStream_t stream) {
  (void)in_sizes; (void)n_in; (void)out_size; (void)ws_size;
  const float* X    = (const float*)d_in[0];
  const int*   Y    = (const int*)d_in[1];
  const float* beta = (const float*)d_in[2];
  const float* phi  = (const float*)d_in[3];
  float* out = (float*)d_out;
  float* ws  = (float*)d_ws;                 // [m(512) | S(512) | lib(512) | partials(512)]
  float* partials = ws + 3 * SAMPLES;
  const int np = NSTILES * GRIDY;            // 512 partials

  nb_stats_kernel<<<SAMPLES, 256, 0, stream>>>(X, Y, beta, ws);
  nb_loglik_kernel<<<dim3(NSTILES, GRIDY), 256, 0, stream>>>(X, Y, beta, phi, ws, partials);
  nb_final_kernel<<<1, 256, 0, stream>>>(partials, np, out);
}